// CudaNewtonSchulzInverse_74629351735426
// MI455X (gfx1250) — compile-verified
//
#include <hip/hip_runtime.h>

// Newton-Schulz batched 128x128 f32 inverse on gfx1250 (MI455X).
// One workgroup (8 wave32) per matrix; W, X, T live in LDS (202KB < 320KB/WGP).
// GEMMs run on V_WMMA_F32_16X16X4_F32 (true-f32 matrix pipe).

typedef __attribute__((ext_vector_type(2))) float v2f;
typedef __attribute__((ext_vector_type(8))) float v8f;

#define MATN 128
#define LDP  132   // padded LDS row stride (floats): bank-conflict-free A frags

// A-fragment (16x4, f32): lane = M (0..15), VGPR0/1 = K,K+1; lanes 16..31 use K+2.
__device__ __forceinline__ v2f lds_load_afrag(const float* A, int row0, int k, int lane) {
    const int r  = lane & 15;
    const int kk = k + ((lane >> 4) << 1);
    return *(const v2f*)(A + (row0 + r) * LDP + kk);   // 8B aligned: kk even, LDP even
}

// B-fragment (4x16, f32): lane = N (0..15), VGPR v = row K+v; lanes 16..31 use K+2.
__device__ __forceinline__ v2f lds_load_bfrag(const float* B, int k, int col0, int lane) {
    const int n  = lane & 15;
    const int kb = k + ((lane >> 4) << 1);
    v2f b;
    b.x = B[(kb    ) * LDP + col0 + n];
    b.y = B[(kb + 1) * LDP + col0 + n];
    return b;
}

// One wave computes a 16x128 strip of D = A(128x128) * B(128x128), both in LDS.
__device__ __forceinline__ void mm_strip(const float* __restrict__ A,
                                         const float* __restrict__ B,
                                         int row0, int lane, v8f acc[8]) {
    const v8f z = {};
#pragma unroll
    for (int j = 0; j < 8; ++j) acc[j] = z;
    for (int k = 0; k < MATN; k += 4) {
        v2f a = lds_load_afrag(A, row0, k, lane);
#pragma unroll
        for (int j = 0; j < 8; ++j) {
            v2f b = lds_load_bfrag(B, k, j << 4, lane);
            // (neg_a, A, neg_b, B, c_mod, C, reuse_a, reuse_b)
            acc[j] = __builtin_amdgcn_wmma_f32_16x16x4_f32(
                false, a, false, b, (short)0, acc[j], false, false);
        }
    }
}

__global__ __launch_bounds__(256, 1)
void ns_inverse_kernel(const float* __restrict__ Wg,
                       float* __restrict__ Xg,
                       const int* __restrict__ itersp)
{
    extern __shared__ float lds[];
    float* Ws = lds;                     // W, resident all iterations
    float* Xs = lds + 1 * MATN * LDP;    // current X
    float* Ts = lds + 2 * MATN * LDP;    // T = 2I - W*X
    __shared__ float red[256];
    __shared__ float s_inv;

    const int tid  = threadIdx.x;
    const int wave = tid >> 5;
    const int lane = tid & 31;
    const int row0 = wave << 4;          // this wave's 16-row output strip
    const int num_iters = *itersp;

    const float* Wsrc = Wg + (size_t)blockIdx.x * (MATN * MATN);

    // ---- Stage W into LDS (coalesced b128 loads/stores) ----
    for (int i = tid * 4; i < MATN * MATN; i += 256 * 4) {
        const int r = i >> 7, c = i & (MATN - 1);
        *(float4*)(Ws + r * LDP + c) = *(const float4*)(Wsrc + i);
    }
    __syncthreads();

    // ---- norm1 = max abs col-sum (tids 0..127), norminf = max abs row-sum (128..255) ----
    {
        float s = 0.0f;
        if (tid < 128) {
            const int col = tid;
            for (int r = 0; r < MATN; ++r) s += fabsf(Ws[r * LDP + col]);
        } else {
            const int row = tid - 128;
            for (int c = 0; c < MATN; ++c) s += fabsf(Ws[row * LDP + c]);
        }
        red[tid] = s;
    }
    __syncthreads();
    for (int stride = 64; stride > 0; stride >>= 1) {
        if (tid < stride)
            red[tid] = fmaxf(red[tid], red[tid + stride]);
        else if (tid >= 128 && tid < 128 + stride)
            red[tid] = fmaxf(red[tid], red[tid + stride]);
        __syncthreads();
    }
    if (tid == 0) s_inv = 1.0f / (red[0] * red[128]);
    __syncthreads();

    // ---- X0 = W^T * (1 / (norm1 * norminf)) ----
    {
        const float inv = s_inv;
        for (int i = tid; i < MATN * MATN; i += 256) {
            const int r = i >> 7, c = i & (MATN - 1);
            Xs[r * LDP + c] = Ws[c * LDP + r] * inv;
        }
    }
    __syncthreads();

    // ---- Newton-Schulz iterations ----
    const int n_c = lane & 15;
    const int m0  = row0 + ((lane >> 4) << 3);   // C/D: lanes 16..31 hold M+8
    for (int it = 0; it < num_iters; ++it) {
        v8f acc[8];

        // T = 2I - W @ X  (fused into the store)
        mm_strip(Ws, Xs, row0, lane, acc);
#pragma unroll
        for (int j = 0; j < 8; ++j) {
            const int col = (j << 4) + n_c;
#pragma unroll
            for (int v = 0; v < 8; ++v) {
                const int row = m0 + v;
                float t = -acc[j][v];
                if (row == col) t += 2.0f;
                Ts[row * LDP + col] = t;
            }
        }
        __syncthreads();   // T visible to all waves

        // X = X @ T   (barrier before overwrite: all waves done reading Xs)
        mm_strip(Xs, Ts, row0, lane, acc);
        __syncthreads();
#pragma unroll
        for (int j = 0; j < 8; ++j) {
            const int col = (j << 4) + n_c;
#pragma unroll
            for (int v = 0; v < 8; ++v)
                Xs[(m0 + v) * LDP + col] = acc[j][v];
        }
        __syncthreads();   // new X visible before next iteration
    }

    // ---- Write X back (coalesced b128 stores) ----
    float* Xdst = Xg + (size_t)blockIdx.x * (MATN * MATN);
    for (int i = tid * 4; i < MATN * MATN; i += 256 * 4) {
        const int r = i >> 7, c = i & (MATN - 1);
        *(float4*)(Xdst + i) = *(const float4*)(Xs + r * LDP + c);
    }
}

extern "C" void kernel_launch(void* const* d_in, const int* in_sizes, int n_in,
                              void* d_out, int out_size, void* d_ws, size_t ws_size,
                              hipStream_t stream) {
    const float* W     = (const float*)d_in[0];
    const int*   iters = (const int*)d_in[1];
    float*       out   = (float*)d_out;

    const int nmat = in_sizes[0] / (MATN * MATN);          // 64*16 = 1024 matrices
    const size_t shmem = (size_t)3 * MATN * LDP * sizeof(float);  // 202752 B < 320KB/WGP

    static_assert(3 * MATN * LDP * sizeof(float) <= 320 * 1024, "LDS budget");

    hipFuncSetAttribute((const void*)ns_inverse_kernel,
                        hipFuncAttributeMaxDynamicSharedMemorySize, (int)shmem);

    ns_inverse_kernel<<<nmat, 256, shmem, stream>>>(W, out, iters);
}